// PointsToObjects_76699525972676
// MI455X (gfx1250) — compile-verified
//
#include <hip/hip_runtime.h>
#include <hip/hip_bf16.h>
#include <stdint.h>

// Problem constants (from reference: [32, 84, 128, 128] fp32, top-100, conf 0.1)
#define B_     32
#define CT_    84
#define CLS_   80
#define HWDIM  128
#define HW2_   (HWDIM * HWDIM)        // 16384
#define N_     (CLS_ * HW2_)          // 1310720 scores per batch
#define K_     100
#define NBIN   4096                   // top-12-bit radix bins
#define CANDM  4096                   // candidate cap per batch (also bitonic size)
#define WGPB   64                     // workgroups per batch for streaming passes
#define CHUNK  (N_ / WGPB)            // 20480 elements per workgroup
#define ITER1  (CHUNK / (256 * 4))    // 20 float4 iterations per thread

// Flip to 0 if the gfx1250 assembler rejects the async mnemonics.
#define USE_ASYNC_STAGE 1

// Monotonic float -> uint key (total order matching float compare)
__device__ __forceinline__ unsigned fkey(float f) {
    unsigned u = __float_as_uint(f);
    return u ^ ((u >> 31) ? 0xFFFFFFFFu : 0x80000000u);
}
__device__ __forceinline__ float fromkey(unsigned k) {
    unsigned u = (k & 0x80000000u) ? (k ^ 0x80000000u) : ~k;
    return __uint_as_float(u);
}

// ---------------- Kernel 0: zero histograms + candidate counters ----------------
__global__ void k_zero(unsigned* __restrict__ p, int n) {
    int i = blockIdx.x * blockDim.x + threadIdx.x;
    if (i < n) p[i] = 0u;
}

// ---------------- Kernel 1: per-batch radix histogram (HBM streaming pass) ------
#if USE_ASYNC_STAGE
// CDNA5 path: per-wave double-buffered GLOBAL_LOAD_ASYNC_TO_LDS_B128 staging.
// Each wave stages 32 lanes x 16B = 512B per step; ASYNCcnt is per-wave and
// async loads complete in order, so s_wait_asynccnt <= 1 guarantees the
// current buffer is resident without any workgroup barrier.
__global__ __launch_bounds__(256) void k_hist(const float* __restrict__ src,
                                              unsigned* __restrict__ ghist) {
    __shared__ unsigned h[NBIN];
    __shared__ float stage[2][256 * 4];                 // 2 x 4KB staging
    const int b = blockIdx.x / WGPB;
    const int w = blockIdx.x % WGPB;
    for (unsigned i = threadIdx.x; i < NBIN; i += 256u) h[i] = 0u;
    __syncthreads();

    const unsigned lane = threadIdx.x & 31u;
    const unsigned wv   = threadIdx.x >> 5;             // 8 waves / block
    const int STEPS = CHUNK / (8 * 128);                // 20 steps of 128 floats/wave
    const float* gbase = src + (size_t)b * CT_ * HW2_ + (size_t)w * CHUNK
                             + (size_t)wv * (CHUNK / 8);

    float* slot0 = &stage[0][threadIdx.x * 4];
    float* slot1 = &stage[1][threadIdx.x * 4];
    const unsigned lds0 = (unsigned)(unsigned long long)slot0;   // LDS byte addr
    const unsigned lds1 = (unsigned)(unsigned long long)slot1;

    {   // prologue: stage step 0 into buffer 0
        unsigned long long g = (unsigned long long)(gbase + lane * 4);
        asm volatile("global_load_async_to_lds_b128 %0, %1, off"
                     :: "v"(lds0), "v"(g) : "memory");
    }
    for (int s = 0; s < STEPS; ++s) {
        if (s + 1 < STEPS) {                            // stage next, wait current
            unsigned long long g =
                (unsigned long long)(gbase + (size_t)(s + 1) * 128 + lane * 4);
            asm volatile("global_load_async_to_lds_b128 %0, %1, off"
                         :: "v"(((s + 1) & 1) ? lds1 : lds0), "v"(g) : "memory");
            asm volatile("s_wait_asynccnt 1" ::: "memory");
        } else {
            asm volatile("s_wait_asynccnt 0" ::: "memory");
        }
        const float4 v = *(const float4*)((s & 1) ? slot1 : slot0);  // ds_load_b128
        atomicAdd(&h[fkey(v.x) >> 20], 1u);             // ds_add
        atomicAdd(&h[fkey(v.y) >> 20], 1u);
        atomicAdd(&h[fkey(v.z) >> 20], 1u);
        atomicAdd(&h[fkey(v.w) >> 20], 1u);
    }
    __syncthreads();

    unsigned* gh = ghist + (size_t)b * NBIN;
    for (unsigned i = threadIdx.x; i < NBIN; i += 256u) {
        unsigned vv = h[i];
        if (vv) atomicAdd(&gh[i], vv);
    }
}
#else
__global__ __launch_bounds__(256) void k_hist(const float* __restrict__ src,
                                              unsigned* __restrict__ ghist) {
    __shared__ unsigned h[NBIN];
    const int b = blockIdx.x / WGPB;
    const int w = blockIdx.x % WGPB;
    for (unsigned i = threadIdx.x; i < NBIN; i += 256u) h[i] = 0u;
    __syncthreads();

    const float4* p = (const float4*)(src + (size_t)b * CT_ * HW2_) + (size_t)w * (CHUNK / 4);
    for (int it = 0; it < ITER1; ++it) {
        if (it + 2 < ITER1)
            __builtin_prefetch(&p[(size_t)(it + 2) * 256 + threadIdx.x], 0, 0);
        float4 v = p[(size_t)it * 256 + threadIdx.x];
        atomicAdd(&h[fkey(v.x) >> 20], 1u);
        atomicAdd(&h[fkey(v.y) >> 20], 1u);
        atomicAdd(&h[fkey(v.z) >> 20], 1u);
        atomicAdd(&h[fkey(v.w) >> 20], 1u);
    }
    __syncthreads();

    unsigned* gh = ghist + (size_t)b * NBIN;
    for (unsigned i = threadIdx.x; i < NBIN; i += 256u) {
        unsigned v = h[i];
        if (v) atomicAdd(&gh[i], v);
    }
}
#endif

// ---------------- Kernel 2: find threshold bin T per batch ----------------------
__global__ __launch_bounds__(256) void k_thresh(const unsigned* __restrict__ ghist,
                                                unsigned* __restrict__ thr) {
    __shared__ unsigned gsum[256];
    const int b = blockIdx.x;
    const unsigned* h = ghist + (size_t)b * NBIN;
    const unsigned t = threadIdx.x;           // each thread sums 16 bins
    unsigned acc = 0;
    for (int i = 0; i < 16; ++i) acc += h[t * 16 + i];
    gsum[t] = acc;
    __syncthreads();

    if (t == 0) {
        unsigned total = 0;
        int g;
        for (g = 255; g >= 0; --g) {          // suffix over 16-bin groups from top
            if (total + gsum[g] >= (unsigned)K_) break;
            total += gsum[g];
        }
        unsigned T = 0;
        if (g >= 0) {
            T = (unsigned)g * 16u;
            unsigned tt = total;
            for (int i = 15; i >= 0; --i) {   // refine inside group g
                unsigned c = h[g * 16 + i];
                if (tt + c >= (unsigned)K_) { T = (unsigned)(g * 16 + i); break; }
                tt += c;
            }
        }
        thr[b] = T;
    }
}

// ---------------- Kernel 3: collect candidates with bin >= T (L2-resident) ------
__global__ __launch_bounds__(256) void k_collect(const float* __restrict__ src,
                                                 const unsigned* __restrict__ thr,
                                                 unsigned* __restrict__ cnt,
                                                 uint2* __restrict__ cand) {
    const int b = blockIdx.x / WGPB;
    const int w = blockIdx.x % WGPB;
    const unsigned T = thr[b];
    const float4* p = (const float4*)(src + (size_t)b * CT_ * HW2_) + (size_t)w * (CHUNK / 4);
    const unsigned baseIdx = (unsigned)w * (unsigned)CHUNK;
    unsigned* c = cnt + b;
    uint2* cb = cand + (size_t)b * CANDM;

    for (int it = 0; it < ITER1; ++it) {
        if (it + 2 < ITER1)
            __builtin_prefetch(&p[(size_t)(it + 2) * 256 + threadIdx.x], 0, 0);
        const unsigned off = (unsigned)it * 256u + threadIdx.x;
        float4 v = p[off];
        const unsigned fi = baseIdx + off * 4u;
        float vals[4] = {v.x, v.y, v.z, v.w};
#pragma unroll
        for (int q = 0; q < 4; ++q) {
            unsigned key = fkey(vals[q]);
            if ((key >> 20) >= T) {
                unsigned pos = atomicAdd(c, 1u);
                if (pos < (unsigned)CANDM) cb[pos] = make_uint2(key, fi + (unsigned)q);
            }
        }
    }
}

// ---------------- Kernel 4: bitonic top-K select + decode + gather + mask -------
__global__ __launch_bounds__(1024) void k_final(const float* __restrict__ src,
                                                const unsigned* __restrict__ cnt,
                                                const uint2* __restrict__ cand,
                                                float* __restrict__ out) {
    __shared__ unsigned long long sk[CANDM];  // 32 KB of the 320 KB WGP LDS
    const int b = blockIdx.x;
    unsigned count = cnt[b];
    if (count > (unsigned)CANDM) count = (unsigned)CANDM;
    const uint2* cb = cand + (size_t)b * CANDM;

    for (unsigned i = threadIdx.x; i < (unsigned)CANDM; i += blockDim.x) {
        unsigned long long v = 0ull;
        if (i < count) {
            uint2 c = cb[i];
            // sort key: score-key major, ~idx minor -> ties resolve to lowest index
            v = ((unsigned long long)c.x << 32) | (unsigned)(~c.y);
        }
        sk[i] = v;
    }
    __syncthreads();

    // bitonic sort, descending
    for (unsigned k = 2; k <= (unsigned)CANDM; k <<= 1) {
        for (unsigned j = k >> 1; j > 0; j >>= 1) {
            for (unsigned i = threadIdx.x; i < (unsigned)CANDM; i += blockDim.x) {
                unsigned ixj = i ^ j;
                if (ixj > i) {
                    unsigned long long a = sk[i], c = sk[ixj];
                    bool desc = ((i & k) == 0);
                    if (desc ? (a < c) : (a > c)) { sk[i] = c; sk[ixj] = a; }
                }
            }
            __syncthreads();
        }
    }

    if (threadIdx.x < (unsigned)K_) {
        unsigned long long e = sk[threadIdx.x];
        unsigned key = (unsigned)(e >> 32);
        unsigned idx = ~(unsigned)e;
        float score = fromkey(key);
        float r0 = 0.f, r1 = 0.f, r2 = 0.f, r3 = 0.f, r4 = 0.f, r5 = 0.f;
        if (threadIdx.x < count && score > 0.1f) {
            unsigned cls = idx >> 14;             // / HW2_
            unsigned sp  = idx & (HW2_ - 1);
            unsigned y   = sp >> 7;
            unsigned x   = sp & (HWDIM - 1);
            const float* eb = src + (size_t)b * CT_ * HW2_ + (size_t)CLS_ * HW2_ + sp;
            r0 = (float)y + eb[0];                // y + offset_y
            r1 = (float)x + eb[HW2_];             // x + offset_x
            r2 = eb[2 * HW2_];                    // height
            r3 = eb[3 * HW2_];                    // width
            r4 = (float)cls;                      // class id
            r5 = score;                           // confidence
        }
        float* o = out + ((size_t)b * K_ + threadIdx.x) * 6;
        o[0] = r0; o[1] = r1; o[2] = r2; o[3] = r3; o[4] = r4; o[5] = r5;
    }
}

extern "C" void kernel_launch(void* const* d_in, const int* in_sizes, int n_in,
                              void* d_out, int out_size, void* d_ws, size_t ws_size,
                              hipStream_t stream) {
    (void)in_sizes; (void)n_in; (void)out_size; (void)ws_size;
    const float* src = (const float*)d_in[0];
    float* out = (float*)d_out;

    // workspace layout (all re-zeroed each call -> deterministic & replay-safe)
    unsigned* hist = (unsigned*)d_ws;                 // 32*4096 u32  = 512 KB
    unsigned* cnt  = hist + (size_t)B_ * NBIN;        // 32 u32
    unsigned* thr  = cnt + B_;                        // 32 u32
    uint2*    cand = (uint2*)(thr + B_);              // 32*4096 u64  = 1 MB

    const int nz = B_ * NBIN + B_;                    // hist + cnt
    hipLaunchKernelGGL(k_zero,    dim3((nz + 255) / 256), dim3(256),  0, stream, hist, nz);
    hipLaunchKernelGGL(k_hist,    dim3(B_ * WGPB),        dim3(256),  0, stream, src, hist);
    hipLaunchKernelGGL(k_thresh,  dim3(B_),               dim3(256),  0, stream, hist, thr);
    hipLaunchKernelGGL(k_collect, dim3(B_ * WGPB),        dim3(256),  0, stream, src, thr, cnt, cand);
    hipLaunchKernelGGL(k_final,   dim3(B_),               dim3(1024), 0, stream, src, cnt, cand, out);
}